// NcmClassifier_65549790871957
// MI455X (gfx1250) — compile-verified
//
#include <hip/hip_runtime.h>
#include <hip/hip_bf16.h>

typedef __attribute__((ext_vector_type(2))) float v2f;
typedef __attribute__((ext_vector_type(8))) float v8f;

// ---------------------------------------------------------------------------
// Prep: rows of `src` [rows, D=64] get scaled by 1/T elementwise into `dst`,
// and per-row squared norms are written to `norms`. One wave32 per row:
// each lane handles 2 consecutive elements (64 = 32 lanes * 2).
// ---------------------------------------------------------------------------
__global__ __launch_bounds__(256) void ncm_prep_kernel(const float* __restrict__ src,
                                                       const float* __restrict__ T,
                                                       float* __restrict__ dst,
                                                       float* __restrict__ norms,
                                                       int rows) {
    const int wave = threadIdx.x >> 5;
    const int lane = threadIdx.x & 31;
    const int row  = blockIdx.x * 8 + wave;
    if (row >= rows) return;

    const v2f s = *(const v2f*)(src + row * 64 + 2 * lane);
    const v2f t = *(const v2f*)(T + 2 * lane);
    float a = s.x / t.x;
    float b = s.y / t.y;
    v2f o; o.x = a; o.y = b;
    *(v2f*)(dst + row * 64 + 2 * lane) = o;

    float ss = a * a + b * b;
#pragma unroll
    for (int off = 16; off > 0; off >>= 1)
        ss += __shfl_xor(ss, off, 32);
    if (lane == 0) norms[row] = ss;
}

// ---------------------------------------------------------------------------
// Main GEMM + epilogue. Each wave computes a 16x64 output strip of
// G = xs @ cs^T via V_WMMA_F32_16X16X4_F32 (full f32 precision, K=4 steps),
// then fuses out = -0.5*sqrt(max(xn + cn - 2G, 0)).
//
// A-tile (16x4 f32, 2 VGPRs), per ISA 7.12.2:
//   lane l<16  : v0 = A[l][k+0], v1 = A[l][k+1]
//   lane l>=16 : v0 = A[l-16][k+2], v1 = A[l-16][k+3]
// B-tile (4x16) mirrors this with the centroid row as the N index
// (B = cs^T, so B[k][n] = cs[n][k], row-major loads work symmetrically).
//
// C/D layout: VGPR r, lanes 0-15 -> (M=r,   N=lane),
//                      lanes 16-31 -> (M=r+8, N=lane-16).
// ---------------------------------------------------------------------------
__global__ __launch_bounds__(256) void ncm_wmma_kernel(const float* __restrict__ xs,
                                                       const float* __restrict__ cs,
                                                       const float* __restrict__ xn,
                                                       const float* __restrict__ cn,
                                                       float* __restrict__ out,
                                                       int N, int M, int D) {
    const int wave  = threadIdx.x >> 5;
    const int lane  = threadIdx.x & 31;
    const int lmod  = lane & 15;
    const int lhalf = lane >> 4;   // 0: K pair (0,1), 1: K pair (2,3)

    const int row0 = blockIdx.y * 128 + wave * 16;   // 16 rows of x per wave
    const int col0 = blockIdx.x * 64;                // 64 centroids per wave

    // Per-lane base pointers (K offset folded in).
    const float* aptr = xs + (size_t)(row0 + lmod) * D + 2 * lhalf;
    const float* b0   = cs + (size_t)(col0 + 0  + lmod) * D + 2 * lhalf;
    const float* b1   = cs + (size_t)(col0 + 16 + lmod) * D + 2 * lhalf;
    const float* b2   = cs + (size_t)(col0 + 32 + lmod) * D + 2 * lhalf;
    const float* b3   = cs + (size_t)(col0 + 48 + lmod) * D + 2 * lhalf;

    v8f acc0 = {}, acc1 = {}, acc2 = {}, acc3 = {};

#pragma unroll
    for (int k = 0; k < 64; k += 4) {
        const v2f a  = *(const v2f*)(aptr + k);
        const v2f vb0 = *(const v2f*)(b0 + k);
        const v2f vb1 = *(const v2f*)(b1 + k);
        const v2f vb2 = *(const v2f*)(b2 + k);
        const v2f vb3 = *(const v2f*)(b3 + k);
        acc0 = __builtin_amdgcn_wmma_f32_16x16x4_f32(false, a, false, vb0,
                                                     (short)0, acc0, false, false);
        acc1 = __builtin_amdgcn_wmma_f32_16x16x4_f32(false, a, false, vb1,
                                                     (short)0, acc1, false, false);
        acc2 = __builtin_amdgcn_wmma_f32_16x16x4_f32(false, a, false, vb2,
                                                     (short)0, acc2, false, false);
        acc3 = __builtin_amdgcn_wmma_f32_16x16x4_f32(false, a, false, vb3,
                                                     (short)0, acc3, false, false);
    }

    // Epilogue: rows this lane owns are mbase..mbase+7.
    const int mbase = row0 + 8 * lhalf;
    float xnv[8];
#pragma unroll
    for (int r = 0; r < 8; ++r) xnv[r] = xn[mbase + r];

    const int ncol0 = col0 + lmod;
    v8f accs[4] = {acc0, acc1, acc2, acc3};
#pragma unroll
    for (int t = 0; t < 4; ++t) {
        const int n   = ncol0 + 16 * t;
        const float c = cn[n];
#pragma unroll
        for (int r = 0; r < 8; ++r) {
            float sq = xnv[r] + c - 2.0f * accs[t][r];
            sq = fmaxf(sq, 0.0f);
            out[(size_t)(mbase + r) * M + n] = -0.5f * __builtin_sqrtf(sq);
        }
    }
}

extern "C" void kernel_launch(void* const* d_in, const int* in_sizes, int n_in,
                              void* d_out, int out_size, void* d_ws, size_t ws_size,
                              hipStream_t stream) {
    const float* x = (const float*)d_in[0];   // [N, D]
    const float* c = (const float*)d_in[1];   // [M, D]
    const float* T = (const float*)d_in[2];   // [D]
    float* out = (float*)d_out;               // [N, M]

    const int D = in_sizes[2];                // 64
    const int N = in_sizes[0] / D;            // 16384
    const int M = in_sizes[1] / D;            // 1024

    // Workspace layout: xs[N*D] | cs[M*D] | xnorm[N] | cnorm[M]
    float* xs = (float*)d_ws;
    float* cs = xs + (size_t)N * D;
    float* xn = cs + (size_t)M * D;
    float* cn = xn + N;

    ncm_prep_kernel<<<dim3((N + 7) / 8), dim3(256), 0, stream>>>(x, T, xs, xn, N);
    ncm_prep_kernel<<<dim3((M + 7) / 8), dim3(256), 0, stream>>>(c, T, cs, cn, M);

    dim3 grid(M / 64, N / 128);   // 16 x 128 blocks
    ncm_wmma_kernel<<<grid, dim3(256), 0, stream>>>(xs, cs, xn, cn, out, N, M, D);
}